// RoiPoolingConv_79345225826357
// MI455X (gfx1250) — compile-verified
//
#include <hip/hip_runtime.h>

// ROI bilinear pooling (tf.image.resize half-pixel centers), POOL=7,
// img (1,64,64,1024) f32, rois (1,1024,4) i32 -> out (1,1024,7,7,1024) f32.
//
// MI455X reasoning: output stream = 205.5 MB -> HBM-store bound (~9us @ 23.3TB/s).
// Image (16.7MB) stays resident in the 192MB L2; NT stores keep the output
// stream from evicting it. The y-direction lerp is D = Wy * Tx via chained
// V_WMMA_F32_16X16X4_F32 (K=16 in 4 steps) on the matrix pipe.
//
// vs previous round: B rows j=14,15 are now CLAMPED to j=13 instead of
// branch-guarded zero-fill. A[:,14..15] == 0 and the clamped data is finite,
// so those products contribute exactly 0 — this deletes the divergent
// s_and_saveexec/cndmask sequences around the gathers.

#define ROI_POOL 7
#define IMG_W 64
#define IMG_C 1024

typedef __attribute__((ext_vector_type(2))) float v2f;
typedef __attribute__((ext_vector_type(8))) float v8f;

// Matches reference exactly:
// scale = size/7; src = (p+0.5)*scale - 0.5; f = floor(src); t = src - f;
// lo = start + clip(f,0,size-1); hi = start + clip(f+1,0,size-1)
__device__ __forceinline__ void axis_coords(int start, int size, int p,
                                            int& lo, int& hi, float& t) {
  float scale = (float)size / 7.0f;
  float src = ((float)p + 0.5f) * scale - 0.5f;
  float f = floorf(src);
  t = src - f;
  int fi = (int)f;
  int l = fi;     if (l < 0) l = 0;       if (l > size - 1) l = size - 1;
  int h = fi + 1; if (h < 0) h = 0;       if (h > size - 1) h = size - 1;
  lo = start + l;
  hi = start + h;
}

__global__ void __launch_bounds__(256)
roi_pool_wmma_kernel(const float* __restrict__ img, const int* __restrict__ rois,
                     float* __restrict__ out, int R) {
  const int bid = blockIdx.x;
  const int roi = bid / ROI_POOL;
  const int px  = bid % ROI_POOL;
  if (roi >= R) return;  // grid is exact; never taken (EXEC stays all-ones)

  const int lane = threadIdx.x & 31;
  const int wave = threadIdx.x >> 5;
  const int lh   = lane >> 4;   // half-wave select: K += 2*lh for A/B layout
  const int ln   = lane & 15;   // A row (M=py) / B,D column (N=channel)

  const int rx = rois[roi * 4 + 0];
  const int ry = rois[roi * 4 + 1];
  const int rw = rois[roi * 4 + 2];
  const int rh = rois[roi * 4 + 3];

  // x-axis coords: uniform across the block (fixed px).
  int gx0, gx1; float tx;
  axis_coords(rx, rw, px, gx0, gx1, tx);
  const float wx0 = 1.0f - tx;

  // ---- A matrix (Wy, 16x16 padded; rows=py, cols=j) as 4 K-slices of 16x4.
  // A element at (lane, vgpr v) holds (M = ln, K = 4q + v + 2*lh).
  // Nonzeros: A[py,py] = 1-ty[py], A[py,py+7] = ty[py], for py < 7.
  float ty_m;
  { int d0, d1; axis_coords(ry, rh, (ln < 7) ? ln : 6, d0, d1, ty_m); }
  v2f A[4];
#pragma unroll
  for (int q = 0; q < 4; ++q) {
#pragma unroll
    for (int v = 0; v < 2; ++v) {
      const int K = 4 * q + v + 2 * lh;
      float aval = 0.0f;
      if (ln < 7) {
        if (K == ln)          aval = 1.0f - ty_m;
        else if (K == ln + 7) aval = ty_m;
      }
      A[q][v] = aval;
    }
  }

  // ---- B gather pointers: B element (lane, v) of slice q holds
  // (K = j = 4q + v + 2*lh, N = channel ln). Row j is the x-lerped image row at
  // yrow[j] = y0[j] (j<7) or y1[j-7]. j >= 14 is clamped to 13: A's columns
  // 14..15 are zero, so the (finite) clamped values contribute nothing —
  // branch-free, no zero-fill, no divergence in the gather path.
  const float* p0base[8];
  const float* p1base[8];
#pragma unroll
  for (int q = 0; q < 4; ++q) {
#pragma unroll
    for (int v = 0; v < 2; ++v) {
      const int e = q * 2 + v;
      int j = 4 * q + v + 2 * lh;
      if (j > 13) j = 13;                     // clamp dead rows to valid data
      const int jm = (j < 7) ? j : j - 7;
      int ylo, yhi; float tdummy;
      axis_coords(ry, rh, jm, ylo, yhi, tdummy);
      const int yr = (j < 7) ? ylo : yhi;
      p0base[e] = img + (yr * IMG_W + gx0) * IMG_C + ln;
      p1base[e] = img + (yr * IMG_W + gx1) * IMG_C + ln;
    }
  }

  // out[(roi*49 + py*7 + px)*1024 + c]
  float* const obase = out + ((size_t)roi * 49 + px) * IMG_C + ln;

  // ---- 8 channel-groups of 16 per wave (8 waves cover all 1024 channels).
#pragma unroll
  for (int i = 0; i < 8; ++i) {
    const int cbase = (wave * 8 + i) * 16;

    v2f B[4];
#pragma unroll
    for (int q = 0; q < 4; ++q) {
#pragma unroll
      for (int v = 0; v < 2; ++v) {
        const int e = q * 2 + v;
        const float p0 = p0base[e][cbase];    // coalesced 64B per half-wave, L2 hit
        const float p1 = p1base[e][cbase];
        B[q][v] = p0 * wx0 + p1 * tx;         // x-direction lerp
      }
    }

    // D = Wy * Tx over K=16 in four K=4 steps on the matrix pipe.
    v8f acc = {0.f, 0.f, 0.f, 0.f, 0.f, 0.f, 0.f, 0.f};
    acc = __builtin_amdgcn_wmma_f32_16x16x4_f32(false, A[0], false, B[0], (short)0, acc, false, false);
    acc = __builtin_amdgcn_wmma_f32_16x16x4_f32(false, A[1], false, B[1], (short)0, acc, false, false);
    acc = __builtin_amdgcn_wmma_f32_16x16x4_f32(false, A[2], false, B[2], (short)0, acc, false, false);
    acc = __builtin_amdgcn_wmma_f32_16x16x4_f32(false, A[3], false, B[3], (short)0, acc, false, false);

    // D rows 0..6 live in vgprs 0..6, lanes 0..15 (M = v + 8*lh).
    if (lane < 16) {
#pragma unroll
      for (int py = 0; py < 7; ++py) {
        __builtin_nontemporal_store(acc[py],
                                    obase + (size_t)py * (ROI_POOL * IMG_C) + cbase);
      }
    }
  }
}

extern "C" void kernel_launch(void* const* d_in, const int* in_sizes, int n_in,
                              void* d_out, int out_size, void* d_ws, size_t ws_size,
                              hipStream_t stream) {
  const float* img  = (const float*)d_in[0];
  const int*   rois = (const int*)d_in[1];
  float*       out  = (float*)d_out;
  const int R = in_sizes[1] / 4;  // 1024 ROIs

  dim3 grid(R * ROI_POOL);        // one block per (roi, px)
  dim3 block(256);                // 8 waves; wave w owns channel groups w*8..w*8+7
  hipLaunchKernelGGL(roi_pool_wmma_kernel, grid, block, 0, stream, img, rois, out, R);
}